// GraphGATClassifierStats_6820408066566
// MI455X (gfx1250) — compile-verified
//
#include <hip/hip_runtime.h>

typedef __attribute__((ext_vector_type(2))) float v2f;
typedef __attribute__((ext_vector_type(8))) float v8f;

#define HEADS 4
#define HID   32
#define HC    128   // HEADS*HID == IN_C
#define NF    15
#define NEG_SLOPE 0.2f

#define LPAD  130   // LDS row stride (floats): bank-conflict-free, keeps 8B alignment for even K
#define TPB_TILES 5 // M-tiles per block (3125 = 625 * 5)

// ---------------- helpers ----------------

__device__ __forceinline__ void atomicMaxF(float* addr, float v) {
  // monotonic int-bit trick: works for mixed signs given -inf init
  if (v >= 0.f) atomicMax((int*)addr, __float_as_int(v));
  else          atomicMin((unsigned int*)addr, __float_as_uint(v));
}

__global__ void fill_kernel(float* __restrict__ p, float v, int n) {
  int i = blockIdx.x * blockDim.x + threadIdx.x;
  if (i < n) p[i] = v;
}

// ---------------- GEMM: C[Mx128] = A[Mx128] * W[128x128], M % (16*TPB_TILES) == 0 -------
// Block = 256 threads = 8 waves. W is staged TRANSPOSED+padded in LDS once per block so
// both A and B fragments are adjacent-pair (8B aligned) DS loads; block then sweeps
// TPB_TILES 16-row M-tiles. Wave w owns output cols [16w, 16w+16). 32 WMMA per tile.

__global__ __launch_bounds__(256) void gemm128_wmma(const float* __restrict__ A,
                                                    const float* __restrict__ W,
                                                    float* __restrict__ C, int mtiles) {
  __shared__ float WT[128 * LPAD];  // WT[n*LPAD + k] = W[k*128 + n]
  __shared__ float As[16 * LPAD];   // As[m*LPAD + k]

  // stage W transposed: coalesced global read; LDS write bank-stride 2 (conflict-free)
  for (int i = threadIdx.x; i < 128 * 128; i += 256) {
    int k = i >> 7, n = i & 127;
    WT[n * LPAD + k] = W[i];
  }

  const int wave  = threadIdx.x >> 5;
  const int lane  = threadIdx.x & 31;
  const int m     = lane & 15;              // A row within tile
  const int khalf = (lane >> 4) << 1;       // lanes 0-15 -> K+0/1, lanes 16-31 -> K+2/3
  const int n     = (wave << 4) + (lane & 15);
  const int rb    = (lane >> 4) << 3;       // lanes 16-31 hold C rows M+8..M+15

  for (int t = 0; t < TPB_TILES; ++t) {
    const int tile = blockIdx.x * TPB_TILES + t;   // uniform across block
    if (tile >= mtiles) break;
    __syncthreads();  // previous tile's readers done (also fences WT staging on t==0)

    const float* Ab = A + (size_t)tile * (16 * 128);
    for (int i = threadIdx.x; i < 16 * 128; i += 256)
      As[(i >> 7) * LPAD + (i & 127)] = Ab[i];
    __syncthreads();

    v8f acc = {};
#pragma unroll
    for (int k0 = 0; k0 < 128; k0 += 4) {
      v2f a, b;
      a[0] = As[m * LPAD + k0 + khalf];
      a[1] = As[m * LPAD + k0 + khalf + 1];
      b[0] = WT[n * LPAD + k0 + khalf];
      b[1] = WT[n * LPAD + k0 + khalf + 1];
      acc = __builtin_amdgcn_wmma_f32_16x16x4_f32(false, a, false, b, (short)0, acc,
                                                  false, false);
    }

    float* Cb = C + (size_t)tile * (16 * 128);
#pragma unroll
    for (int r = 0; r < 8; ++r) Cb[(r + rb) * 128 + n] = acc[r];
  }
}

// ---------------- attention scores: a_src/a_dst [N,H] ----------------
// one wave32 per node; lane = channel within head; shuffle reduction.

__global__ void attn_scores(const float* __restrict__ xh,
                            const float* __restrict__ att_s,
                            const float* __restrict__ att_d,
                            float* __restrict__ asrc, float* __restrict__ adst, int Nn) {
  int gid = blockIdx.x * blockDim.x + threadIdx.x;
  int n = gid >> 5, lane = gid & 31;
  if (n >= Nn) return;
#pragma unroll
  for (int h = 0; h < HEADS; ++h) {
    float v  = xh[(size_t)n * HC + h * HID + lane];
    float ps = v * att_s[h * HID + lane];
    float pd = v * att_d[h * HID + lane];
#pragma unroll
    for (int off = 16; off > 0; off >>= 1) {
      ps += __shfl_xor(ps, off, 32);
      pd += __shfl_xor(pd, off, 32);
    }
    if (lane == 0) {
      asrc[n * HEADS + h] = ps;
      adst[n * HEADS + h] = pd;
    }
  }
}

// ---------------- edge pass 1: e = leaky_relu(asrc[s]+adst[d]); segment max ----------------

__global__ void edge_pass1(const int* __restrict__ ei, int E, int Nn,
                           const float* __restrict__ asrc, const float* __restrict__ adst,
                           float* __restrict__ ebuf, float* __restrict__ emax) {
  int e = blockIdx.x * blockDim.x + threadIdx.x;
  if (e >= E + Nn) return;
  int s = (e < E) ? ei[e]     : (e - E);
  int d = (e < E) ? ei[E + e] : (e - E);
  const float4 va = *(const float4*)&asrc[(size_t)s * HEADS];
  const float4 vb = *(const float4*)&adst[(size_t)d * HEADS];
  float v[HEADS] = {va.x + vb.x, va.y + vb.y, va.z + vb.z, va.w + vb.w};
#pragma unroll
  for (int h = 0; h < HEADS; ++h) {
    v[h] = (v[h] > 0.f) ? v[h] : NEG_SLOPE * v[h];
    atomicMaxF(&emax[d * HEADS + h], v[h]);
  }
  *(float4*)&ebuf[(size_t)e * HEADS] = make_float4(v[0], v[1], v[2], v[3]);
}

// ---------------- edge pass 2: ex = exp(e - emax[d]); segment sum ----------------

__global__ void edge_pass2(const int* __restrict__ ei, int E, int Nn,
                           const float* __restrict__ emax,
                           float* __restrict__ ebuf, float* __restrict__ denom) {
  int e = blockIdx.x * blockDim.x + threadIdx.x;
  if (e >= E + Nn) return;
  int d = (e < E) ? ei[E + e] : (e - E);
  const float4 ev = *(const float4*)&ebuf[(size_t)e * HEADS];
  const float4 mv = *(const float4*)&emax[(size_t)d * HEADS];
  float v[HEADS] = {__expf(ev.x - mv.x), __expf(ev.y - mv.y),
                    __expf(ev.z - mv.z), __expf(ev.w - mv.w)};
#pragma unroll
  for (int h = 0; h < HEADS; ++h) atomicAdd(&denom[d * HEADS + h], v[h]);
  *(float4*)&ebuf[(size_t)e * HEADS] = make_float4(v[0], v[1], v[2], v[3]);
}

// ---------------- edge pass 3: out[d] += alpha * xh[s], wave per edge ----------------
// lane = channel within head -> 32-lane coalesced global_atomic_add_f32.

__global__ void edge_pass3(const int* __restrict__ ei, int E, int Nn,
                           const float* __restrict__ ebuf,
                           const float* __restrict__ denom,
                           const float* __restrict__ xh, float* __restrict__ out) {
  int gid  = blockIdx.x * blockDim.x + threadIdx.x;
  int e    = gid >> 5;
  int lane = gid & 31;
  if (e >= E + Nn) return;
  int s = (e < E) ? ei[e]     : (e - E);
  int d = (e < E) ? ei[E + e] : (e - E);
  const float4 ev = *(const float4*)&ebuf[(size_t)e * HEADS];
  const float4 dv = *(const float4*)&denom[(size_t)d * HEADS];
  float alpha[HEADS] = {ev.x / (dv.x + 1e-16f), ev.y / (dv.y + 1e-16f),
                        ev.z / (dv.z + 1e-16f), ev.w / (dv.w + 1e-16f)};
#pragma unroll
  for (int h = 0; h < HEADS; ++h) {
    atomicAdd(&out[(size_t)d * HC + h * HID + lane],
              alpha[h] * xh[(size_t)s * HC + h * HID + lane]);
  }
}

// ---------------- h = relu(out + b1), in place ----------------

__global__ void bias_relu(float* __restrict__ h, const float* __restrict__ b, int total) {
  int i = blockIdx.x * blockDim.x + threadIdx.x;
  if (i >= total) return;
  float v = h[i] + b[i & (HC - 1)];
  h[i] = (v > 0.f) ? v : 0.f;
}

// ---------------- head mean + bias, fused with batch mean-pool accumulation ----------------

__global__ void head_mean_pool(const float* __restrict__ out2, const float* __restrict__ b2,
                               const int* __restrict__ batch,
                               float* __restrict__ gsum, float* __restrict__ cnt, int Nn) {
  int i = blockIdx.x * blockDim.x + threadIdx.x;
  if (i >= Nn * HID) return;
  int n = i >> 5, c = i & 31;
  float m = 0.f;
#pragma unroll
  for (int h = 0; h < HEADS; ++h) m += out2[(size_t)n * HC + h * HID + c];
  m = m * (1.0f / HEADS) + b2[c];
  int g = batch[n];
  atomicAdd(&gsum[g * HID + c], m);
  if (c == 0) atomicAdd(&cnt[g], 1.0f);
}

// ---------------- final MLP: [gmean | stats](47) -> relu(32) -> 1 ----------------

__global__ void mlp_kernel(const float* __restrict__ gsum, const float* __restrict__ cnt,
                           const float* __restrict__ stats,
                           const float* __restrict__ Wm1, const float* __restrict__ bm1,
                           const float* __restrict__ Wm2, const float* __restrict__ bm2,
                           float* __restrict__ out, int Bn) {
  int b = blockIdx.x * blockDim.x + threadIdx.x;
  if (b >= Bn) return;
  float feat[HID + NF];
  float c = cnt[b];
  c = (c > 1.f) ? c : 1.f;
#pragma unroll
  for (int i = 0; i < HID; ++i) feat[i] = gsum[b * HID + i] / c;
#pragma unroll
  for (int i = 0; i < NF; ++i) feat[HID + i] = stats[b * NF + i];
  float o = bm2[0];
  for (int j = 0; j < HID; ++j) {
    float hs = bm1[j];
    for (int i = 0; i < HID + NF; ++i) hs += feat[i] * Wm1[i * HID + j];
    hs = (hs > 0.f) ? hs : 0.f;
    o += hs * Wm2[j];
  }
  out[b] = o;
}

// ---------------- launch ----------------

extern "C" void kernel_launch(void* const* d_in, const int* in_sizes, int n_in,
                              void* d_out, int out_size, void* d_ws, size_t ws_size,
                              hipStream_t stream) {
  const float* x     = (const float*)d_in[0];
  const int*   ei    = (const int*)  d_in[1];
  const int*   batch = (const int*)  d_in[2];
  const float* stats = (const float*)d_in[3];
  const float* W1    = (const float*)d_in[4];
  const float* as1   = (const float*)d_in[5];
  const float* ad1   = (const float*)d_in[6];
  const float* b1    = (const float*)d_in[7];
  const float* W2    = (const float*)d_in[8];
  const float* as2   = (const float*)d_in[9];
  const float* ad2   = (const float*)d_in[10];
  const float* b2    = (const float*)d_in[11];
  const float* Wm1   = (const float*)d_in[12];
  const float* bm1   = (const float*)d_in[13];
  const float* Wm2   = (const float*)d_in[14];
  const float* bm2   = (const float*)d_in[15];

  const int Nn = in_sizes[0] / HC;   // 50000
  const int E  = in_sizes[1] / 2;    // 800000
  const int Bn = in_sizes[3] / NF;   // 100
  const int ET = E + Nn;             // edges incl. self loops

  // workspace layout (floats)
  float* ws = (float*)d_ws;
  size_t o = 0;
  float* xh    = ws + o; o += (size_t)Nn * HC;       // GEMM output (both layers)
  float* agg   = ws + o; o += (size_t)Nn * HC;       // aggregation / h1
  float* asrc  = ws + o; o += (size_t)Nn * HEADS;
  float* adst  = ws + o; o += (size_t)Nn * HEADS;
  float* emax  = ws + o; o += (size_t)Nn * HEADS;
  float* denom = ws + o; o += (size_t)Nn * HEADS;
  float* ebuf  = ws + o; o += (size_t)ET * HEADS;
  float* gsum  = ws + o; o += (size_t)Bn * HID;
  float* cnt   = ws + o; o += (size_t)Bn;

  const int mtiles = Nn / 16;                 // 3125
  const int nh     = Nn * HEADS;
  const int nhc    = Nn * HC;
  const int TB = 256;
  auto cdiv = [](int a, int b) { return (a + b - 1) / b; };
  const int gemm_blocks = cdiv(mtiles, TPB_TILES);   // 625

  // ---- layer 1 ----
  gemm128_wmma<<<gemm_blocks, TB, 0, stream>>>(x, W1, xh, mtiles);
  attn_scores<<<cdiv(Nn * 32, TB), TB, 0, stream>>>(xh, as1, ad1, asrc, adst, Nn);
  fill_kernel<<<cdiv(nh, TB), TB, 0, stream>>>(emax, -__builtin_huge_valf(), nh);
  fill_kernel<<<cdiv(nh, TB), TB, 0, stream>>>(denom, 0.f, nh);
  fill_kernel<<<cdiv(nhc, TB), TB, 0, stream>>>(agg, 0.f, nhc);
  edge_pass1<<<cdiv(ET, TB), TB, 0, stream>>>(ei, E, Nn, asrc, adst, ebuf, emax);
  edge_pass2<<<cdiv(ET, TB), TB, 0, stream>>>(ei, E, Nn, emax, ebuf, denom);
  edge_pass3<<<cdiv(ET * 32, TB), TB, 0, stream>>>(ei, E, Nn, ebuf, denom, xh, agg);
  bias_relu<<<cdiv(nhc, TB), TB, 0, stream>>>(agg, b1, nhc);   // agg = h1

  // ---- layer 2 ----
  gemm128_wmma<<<gemm_blocks, TB, 0, stream>>>(agg, W2, xh, mtiles);  // xh = xh2
  attn_scores<<<cdiv(Nn * 32, TB), TB, 0, stream>>>(xh, as2, ad2, asrc, adst, Nn);
  fill_kernel<<<cdiv(nh, TB), TB, 0, stream>>>(emax, -__builtin_huge_valf(), nh);
  fill_kernel<<<cdiv(nh, TB), TB, 0, stream>>>(denom, 0.f, nh);
  fill_kernel<<<cdiv(nhc, TB), TB, 0, stream>>>(agg, 0.f, nhc);
  edge_pass1<<<cdiv(ET, TB), TB, 0, stream>>>(ei, E, Nn, asrc, adst, ebuf, emax);
  edge_pass2<<<cdiv(ET, TB), TB, 0, stream>>>(ei, E, Nn, emax, ebuf, denom);
  edge_pass3<<<cdiv(ET * 32, TB), TB, 0, stream>>>(ei, E, Nn, ebuf, denom, xh, agg);

  // ---- pooling + MLP head ----
  fill_kernel<<<cdiv(Bn * HID, TB), TB, 0, stream>>>(gsum, 0.f, Bn * HID);
  fill_kernel<<<cdiv(Bn, TB), TB, 0, stream>>>(cnt, 0.f, Bn);
  head_mean_pool<<<cdiv(Nn * HID, TB), TB, 0, stream>>>(agg, b2, batch, gsum, cnt, Nn);
  mlp_kernel<<<1, 128, 0, stream>>>(gsum, cnt, stats, Wm1, bm1, Wm2, bm2,
                                    (float*)d_out, Bn);
}